// Top2Gating_75617194213751
// MI455X (gfx1250) — compile-verified
//
#include <hip/hip_runtime.h>

// ---------------------------------------------------------------------------
// Top2Gating for MI455X (gfx1250, wave32).
// x:(T=4,B=32,D=512,H=24,W=24) f32, w:(512,16) f32
// outputs: dispatch(B,N,E,C), combine(B,N,E,C), loss   with N=576,E=16,C=144
//
// Roofline: ~151MB x-read + ~340MB output write  =>  ~21us @ 23.3TB/s.
// GEMM is 0.3 GFLOP -> irrelevant; done exactly in f32 WMMA (16x16x4).
// ---------------------------------------------------------------------------

typedef __attribute__((ext_vector_type(2))) float v2f;
typedef __attribute__((ext_vector_type(4))) float v4f;
typedef __attribute__((ext_vector_type(8))) float v8f;

#define T_    4
#define B_    32
#define D_    512
#define N_    576
#define E_    16
#define CAP_  144
#define CHUNKS_ 18         // N_/32 : 32-token groups per batch image

// ---------------------------------------------------------------------------
// Kernel 1: fused T-pool + GEMM (two 16x16 tiles / wave via V_WMMA_F32_16X16X4_F32)
//           + LDS transpose so each lane owns one token, then scalar
//           softmax/top-2 with no cross-lane ops at all.
// grid: B_*CHUNKS_ blocks, 32 threads (1 wave).
// ---------------------------------------------------------------------------
__global__ __launch_bounds__(32)
void gating_logits_top2(const float* __restrict__ x,
                        const float* __restrict__ w,
                        float* __restrict__ rawws,         // [B_*N_][E_] raw gates
                        int*   __restrict__ e0a,           // [B_*N_]
                        int*   __restrict__ e1a,
                        float* __restrict__ g0a,
                        float* __restrict__ g1a)
{
    const int blk   = blockIdx.x;
    const int b     = blk / CHUNKS_;
    const int chunk = blk % CHUNKS_;
    const int base  = chunk * 32;       // first of this wave's 32 tokens
    const int lane  = threadIdx.x;      // wave32
    const int m     = lane & 15;        // A-matrix row within tile
    const int h     = lane >> 4;        // lane half -> K pair {0,1} vs {2,3}
    const int e     = lane & 15;        // B/C column (expert)

    // x element address: ((t*B + b)*D + d)*N + n ; tokens contiguous in n.
    const float* xrow = x + (size_t)b * D_ * N_ + base + m;
    const size_t tstride = (size_t)B_ * D_ * N_;

    v8f acc0 = {};   // logits for tokens base+0..15
    v8f acc1 = {};   // logits for tokens base+16..31

    for (int d0 = 0; d0 < D_; d0 += 4) {
        const int dA = d0 + 2 * h;                // this half's K values dA, dA+1
        const float* pa = xrow + (size_t)dA * N_;
        __builtin_prefetch(pa + 4 * (size_t)N_, 0, 0);   // next K-block (global_prefetch)
        float s00 = 0.f, s01 = 0.f, s10 = 0.f, s11 = 0.f;
        #pragma unroll
        for (int t = 0; t < T_; ++t) {            // fuse mean over T; x is read-once -> NT
            const float* pt = pa + t * tstride;
            s00 += __builtin_nontemporal_load(pt);             // tile0, K=dA
            s01 += __builtin_nontemporal_load(pt + N_);        // tile0, K=dA+1
            s10 += __builtin_nontemporal_load(pt + 16);        // tile1, K=dA
            s11 += __builtin_nontemporal_load(pt + N_ + 16);   // tile1, K=dA+1
        }
        v2f a0; a0[0] = s00 * 0.25f; a0[1] = s01 * 0.25f;
        v2f a1; a1[0] = s10 * 0.25f; a1[1] = s11 * 0.25f;
        v2f bb; bb[0] = w[(size_t)dA * E_ + e]; bb[1] = w[(size_t)(dA + 1) * E_ + e];
        acc0 = __builtin_amdgcn_wmma_f32_16x16x4_f32(false, a0, false, bb,
                                                     (short)0, acc0, false, false);
        acc1 = __builtin_amdgcn_wmma_f32_16x16x4_f32(false, a1, false, bb,
                                                     (short)0, acc1, false, false);
    }

    // Transpose C fragments through LDS (pad 17: gcd(17,64)=1 -> conflict-free).
    // C layout: VGPR r -> token-row r (lanes 0-15) / r+8 (lanes 16-31), col = lane&15.
    __shared__ float lds[32 * 17];
    #pragma unroll
    for (int r = 0; r < 8; ++r) {
        lds[(r + 8 * h) * 17 + e]      = acc0[r];
        lds[(16 + r + 8 * h) * 17 + e] = acc1[r];
    }
    __syncthreads();

    // Each lane now owns token (base+lane): 16 logits in registers.
    float g[E_];
    #pragma unroll
    for (int j = 0; j < E_; ++j) g[j] = lds[lane * 17 + j];

    float mx = g[0];
    #pragma unroll
    for (int j = 1; j < E_; ++j) mx = fmaxf(mx, g[j]);
    float raw[E_];
    float s = 0.f;
    #pragma unroll
    for (int j = 0; j < E_; ++j) { raw[j] = __expf(g[j] - mx); s += raw[j]; }
    const float invs = 1.f / s;
    #pragma unroll
    for (int j = 0; j < E_; ++j) raw[j] *= invs;        // raw_gates (softmax)

    // top-1 (strict > keeps first occurrence, like jnp.argmax)
    float v1 = raw[0]; int i1 = 0;
    #pragma unroll
    for (int j = 1; j < E_; ++j) if (raw[j] > v1) { v1 = raw[j]; i1 = j; }
    // top-2 on gates with top-1 removed
    float v2 = -1.f; int i2 = 0;
    #pragma unroll
    for (int j = 0; j < E_; ++j) if (j != i1 && raw[j] > v2) { v2 = raw[j]; i2 = j; }

    const float gs = v1 + v2 + 1e-9f;                   // + EPS
    const int tok = b * N_ + base + lane;
    e0a[tok] = i1;        e1a[tok] = i2;                // coalesced per-token writes
    g0a[tok] = v1 / gs;   g1a[tok] = v2 / gs;

    // raw gates for the load-balance loss: 64B contiguous per lane, coalesced.
    v4f* rw = (v4f*)(rawws + (size_t)tok * E_);
    #pragma unroll
    for (int q = 0; q < 4; ++q) {
        v4f v; v[0] = raw[4*q]; v[1] = raw[4*q+1]; v[2] = raw[4*q+2]; v[3] = raw[4*q+3];
        rw[q] = v;
    }
}

// ---------------------------------------------------------------------------
// Kernel 2: per-batch sequential capacity assignment via wave32 ballot scan.
// grid: B_ blocks, 32 threads.
// ---------------------------------------------------------------------------
__global__ __launch_bounds__(32)
void capacity_scan(const float* __restrict__ rawws,
                   const int*   __restrict__ e0a, const int*   __restrict__ e1a,
                   const float* __restrict__ g0a, const float* __restrict__ g1a,
                   int*   __restrict__ p0a, int*   __restrict__ p1a,
                   float* __restrict__ g0f, float* __restrict__ g1f,
                   float* __restrict__ loss_partial)
{
    const int b    = blockIdx.x;
    const int lane = threadIdx.x;
    const unsigned ltmask = (1u << lane) - 1u;

    // pass 1: total top-1 count per expert (pre-capacity): needed for the
    // top-2 base offset min(count1,cap) and for density_1 in the loss.
    int cnt1[E_];
    #pragma unroll
    for (int ee = 0; ee < E_; ++ee) cnt1[ee] = 0;
    for (int c = 0; c < CHUNKS_; ++c) {
        const int e0 = e0a[b * N_ + c * 32 + lane];
        #pragma unroll
        for (int ee = 0; ee < E_; ++ee)
            cnt1[ee] += __popc((unsigned)__ballot(e0 == ee));
    }

    int base1[E_], run0[E_], run1[E_];
    #pragma unroll
    for (int ee = 0; ee < E_; ++ee) {
        base1[ee] = cnt1[ee] < CAP_ ? cnt1[ee] : CAP_;
        run0[ee] = 0; run1[ee] = 0;
    }

    // pass 2: exclusive-cumsum positions in token order via ballot rank
    for (int c = 0; c < CHUNKS_; ++c) {
        const int tok = b * N_ + c * 32 + lane;
        const int   e0 = e0a[tok], e1 = e1a[tok];
        const float g0 = g0a[tok], g1 = g1a[tok];
        int pos0 = 0, pos1 = 0;
        #pragma unroll
        for (int ee = 0; ee < E_; ++ee) {
            const unsigned m0 = (unsigned)__ballot(e0 == ee);
            const unsigned m1 = (unsigned)__ballot(e1 == ee);
            if (e0 == ee) pos0 = run0[ee] + __popc(m0 & ltmask);
            if (e1 == ee) pos1 = base1[ee] + run1[ee] + __popc(m1 & ltmask);
            run0[ee] += __popc(m0);
            run1[ee] += __popc(m1);
        }
        p0a[tok] = pos0;
        p1a[tok] = pos1;
        g0f[tok] = (pos0 < CAP_) ? g0 : 0.f;   // capacity-dropped -> gate 0
        g1f[tok] = (pos1 < CAP_) ? g1 : 0.f;
    }

    // loss partial: sum_e raw_sum[b,e] * count1[b,e]
    float lp = 0.f;
    if (lane < E_) {
        float sr = 0.f;
        for (int n = 0; n < N_; ++n)
            sr += rawws[((size_t)b * N_ + n) * E_ + lane];   // coalesced over lanes
        float myc = 0.f;
        #pragma unroll
        for (int ee = 0; ee < E_; ++ee) if (lane == ee) myc = (float)cnt1[ee];
        lp = sr * myc;
    }
    #pragma unroll
    for (int msk = 1; msk < 32; msk <<= 1) lp += __shfl_xor(lp, msk, 32);
    if (lane == 0) loss_partial[b] = lp;
}

// ---------------------------------------------------------------------------
// Kernel 3: zero-fill output (dominant, unavoidable bandwidth). Output exceeds
// L2 (340MB > 192MB) -> non-temporal streaming stores (ext_vector float4).
// ---------------------------------------------------------------------------
__global__ void zero_out(float* __restrict__ out, long n)
{
    const long i = (long)blockIdx.x * blockDim.x + threadIdx.x;
    const long n4 = n >> 2;
    v4f* o4 = (v4f*)out;
    const v4f z = {0.f, 0.f, 0.f, 0.f};
    if (i < n4) __builtin_nontemporal_store(z, &o4[i]);
    if (i == 0)
        for (long k = n4 << 2; k < n; ++k) out[k] = 0.f;
}

// ---------------------------------------------------------------------------
// Kernel 4: sparse scatter of the 2 nonzeros per token + loss finalize.
// ---------------------------------------------------------------------------
__global__ void scatter_combine(const int*   __restrict__ e0a,
                                const int*   __restrict__ e1a,
                                const int*   __restrict__ p0a,
                                const int*   __restrict__ p1a,
                                const float* __restrict__ g0f,
                                const float* __restrict__ g1f,
                                const float* __restrict__ loss_partial,
                                float* __restrict__ out)
{
    const long NB   = (long)B_ * N_;
    const long SZ   = NB * E_ * CAP_;
    const long tid  = (long)blockIdx.x * blockDim.x + threadIdx.x;
    float* dispatch = out;
    float* combine  = out + SZ;

    if (tid < 2 * NB) {
        const long tok = tid >> 1;
        const int  k   = (int)(tid & 1);
        const int   e = k ? e1a[tok] : e0a[tok];
        const int   p = k ? p1a[tok] : p0a[tok];
        const float g = k ? g1f[tok] : g0f[tok];
        if (g != 0.f && p >= 0 && p < CAP_) {
            const long idx = (tok * E_ + e) * (long)CAP_ + p;
            combine[idx]  = g;
            dispatch[idx] = 1.f;      // dispatch = (combine != 0)
        }
    }
    if (tid == 0) {
        float s = 0.f;
        for (int bb = 0; bb < B_; ++bb) s += loss_partial[bb];
        // loss = E^2 * mean(proxy * density) = s * E / (B * N^2)
        out[2 * SZ] = s * ((float)E_ / ((float)B_ * (float)N_ * (float)N_));
    }
}

// ---------------------------------------------------------------------------
extern "C" void kernel_launch(void* const* d_in, const int* in_sizes, int n_in,
                              void* d_out, int out_size, void* d_ws, size_t ws_size,
                              hipStream_t stream)
{
    const float* x = (const float*)d_in[0];   // (4,32,512,24,24)
    const float* w = (const float*)d_in[1];   // (512,16)
    float* out = (float*)d_out;

    // workspace layout (rawws first: 16B-aligned at base)
    char* ws = (char*)d_ws;
    const size_t NT = (size_t)B_ * N_;
    float* rawws        = (float*)ws;                 ws += NT * E_ * 4;
    int*   e0a          = (int*)ws;                   ws += NT * 4;
    int*   e1a          = (int*)ws;                   ws += NT * 4;
    float* g0a          = (float*)ws;                 ws += NT * 4;
    float* g1a          = (float*)ws;                 ws += NT * 4;
    int*   p0a          = (int*)ws;                   ws += NT * 4;
    int*   p1a          = (int*)ws;                   ws += NT * 4;
    float* g0f          = (float*)ws;                 ws += NT * 4;
    float* g1f          = (float*)ws;                 ws += NT * 4;
    float* loss_partial = (float*)ws;                 ws += B_ * 4;

    // 1) pooled GEMM (WMMA) + softmax + top-2
    gating_logits_top2<<<B_ * CHUNKS_, 32, 0, stream>>>(x, w, rawws,
                                                        e0a, e1a, g0a, g1a);
    // 2) capacity scan (sequential semantics via wave32 ballot ranks)
    capacity_scan<<<B_, 32, 0, stream>>>(rawws, e0a, e1a, g0a, g1a,
                                         p0a, p1a, g0f, g1f, loss_partial);
    // 3) zero the dense outputs
    {
        const long n  = (long)out_size;          // 2*B*N*E*C + 1
        const long n4 = n >> 2;
        const int blocks = (int)((n4 + 255) / 256);
        zero_out<<<blocks, 256, 0, stream>>>(out, n);
    }
    // 4) scatter nonzeros + loss
    {
        const long nth = 2L * B_ * N_;
        const int blocks = (int)((nth + 255) / 256);
        scatter_combine<<<blocks, 256, 0, stream>>>(e0a, e1a, p0a, p1a,
                                                    g0f, g1f, loss_partial, out);
    }
}